// LuongAttention_24086176596734
// MI455X (gfx1250) — compile-verified
//
#include <hip/hip_runtime.h>
#include <stdint.h>

typedef __attribute__((ext_vector_type(2))) float v2f;
typedef __attribute__((ext_vector_type(8))) float v8f;

#define TILE_M 128
#define TILE_N 128
#define TILE_K 16
#define A_LD   20          // 80 B row stride: 16B-aligned, bank-conflict-free

#if __has_builtin(__builtin_amdgcn_s_wait_asynccnt)
#define WAIT_ASYNC(n) __builtin_amdgcn_s_wait_asynccnt(n)
#else
#define WAIT_ASYNC(n) asm volatile("s_wait_asynccnt %0" :: "i"(n) : "memory")
#endif

// One lane copies 16 bytes global -> LDS via the CDNA5 async engine (ASYNCcnt).
__device__ __forceinline__ void async_b128(const float* lds_dst, const float* gsrc) {
  const uint32_t lds_off = (uint32_t)(uintptr_t)lds_dst;   // low 32 bits of a
  // generic shared pointer == wave-relative LDS byte address on AMDGPU.
  asm volatile("global_load_async_to_lds_b128 %0, %1, off"
               :: "v"(lds_off), "v"(gsrc) : "memory");
}

// D = A * B (+bias broadcast over rows), fp32 WMMA 16x16x4, double-buffered
// async-to-LDS pipeline.
// TRANS_B=false: Bm is [K x N] row-major, staged as Bs[k][n] (stride 136).
// TRANS_B=true : Bm is [N x K] row-major, staged as Bs[n][k] (stride 20,
//                same shape as the A tile; B-fragment reads use A's pattern).
// 256 threads = 8 waves; workgroup tile 128x128; each wave owns 32x64 =
// 2x4 V_WMMA_F32_16X16X4_F32 accumulators.
template<bool TRANS_B, bool ADD_BIAS>
__global__ __launch_bounds__(256) void wmma_gemm_f32_kernel(
    const float* __restrict__ A, const float* __restrict__ Bm,
    const float* __restrict__ bias, float* __restrict__ C,
    int K, long long strideA, long long strideB, long long strideC,
    int lda, int ldb, int ldc)
{
  constexpr int B_ROWS = TRANS_B ? TILE_N : TILE_K;
  constexpr int B_LD   = TRANS_B ? A_LD : (TILE_N + 8);    // 20 or 136

  __shared__ float Abuf[2][TILE_M * A_LD];
  __shared__ float Bbuf[2][B_ROWS * B_LD];

  A  += (long long)blockIdx.z * strideA;
  Bm += (long long)blockIdx.z * strideB;
  C  += (long long)blockIdx.z * strideC;

  const int tid  = threadIdx.x;
  const int lane = tid & 31;
  const int wid  = tid >> 5;
  const int wm   = (wid & 3) * 32;     // wave row offset (0,32,64,96)
  const int wn   = (wid >> 2) * 64;    // wave col offset (0,64)
  const int mm   = lane & 15;          // M (A frag) / N (B,C frags) index
  const int kk   = (lane >> 4) * 2;    // K pair: lanes 0-15 -> {0,1}, 16-31 -> {2,3}
  const int hi   = lane >> 4;          // C/D row-half select

  const int tileM = blockIdx.y * TILE_M;
  const int tileN = blockIdx.x * TILE_N;

  // Issue one k-chunk's staging: 4 async b128 per thread (A:2 + B:2) = 4 per wave.
  auto issue = [&](int kc, int buf) {
    #pragma unroll
    for (int i = 0; i < 2; ++i) {                       // A: 128x16 = 512 float4
      const int idx = tid + i * 256;
      const int r = idx >> 2, c = (idx & 3) * 4;
      async_b128(&Abuf[buf][r * A_LD + c],
                 A + (long long)(tileM + r) * lda + kc + c);
    }
    if (TRANS_B) {                                      // B: 128 rows x 16 k
      #pragma unroll
      for (int i = 0; i < 2; ++i) {
        const int idx = tid + i * 256;
        const int r = idx >> 2, c = (idx & 3) * 4;
        async_b128(&Bbuf[buf][r * B_LD + c],
                   Bm + (long long)(tileN + r) * ldb + kc + c);
      }
    } else {                                            // B: 16 k x 128 n
      #pragma unroll
      for (int i = 0; i < 2; ++i) {
        const int idx = tid + i * 256;
        const int r = idx >> 5, c = (idx & 31) * 4;
        async_b128(&Bbuf[buf][r * B_LD + c],
                   Bm + (long long)(kc + r) * ldb + tileN + c);
      }
    }
  };

  v8f acc[2][4] = {};
  const int nchunks = K / TILE_K;

  issue(0, 0);
  for (int ci = 0; ci < nchunks; ++ci) {
    if (ci + 1 < nchunks) {
      issue((ci + 1) * TILE_K, (ci + 1) & 1);  // prefetch next chunk
      WAIT_ASYNC(4);                           // in-order: current chunk landed
    } else {
      WAIT_ASYNC(0);
    }
    __syncthreads();

    const float* As = Abuf[ci & 1];
    const float* Bs = Bbuf[ci & 1];
    #pragma unroll
    for (int k4 = 0; k4 < TILE_K; k4 += 4) {
      v2f a[2], b[4];
      #pragma unroll
      for (int mi = 0; mi < 2; ++mi) {
        a[mi].x = As[(wm + mi * 16 + mm) * A_LD + k4 + kk];
        a[mi].y = As[(wm + mi * 16 + mm) * A_LD + k4 + kk + 1];
      }
      #pragma unroll
      for (int ni = 0; ni < 4; ++ni) {
        if (TRANS_B) {
          b[ni].x = Bs[(wn + ni * 16 + mm) * B_LD + k4 + kk];
          b[ni].y = Bs[(wn + ni * 16 + mm) * B_LD + k4 + kk + 1];
        } else {
          b[ni].x = Bs[(k4 + kk) * B_LD + wn + ni * 16 + mm];
          b[ni].y = Bs[(k4 + kk + 1) * B_LD + wn + ni * 16 + mm];
        }
      }
      #pragma unroll
      for (int mi = 0; mi < 2; ++mi)
        #pragma unroll
        for (int ni = 0; ni < 4; ++ni)
          acc[mi][ni] = __builtin_amdgcn_wmma_f32_16x16x4_f32(
              false, a[mi], false, b[ni], (short)0, acc[mi][ni], false, false);
    }
    __syncthreads();  // all waves done reading this buffer before it is reused
  }

  // Epilogue: D layout = VGPR r -> rows {r, r+8} x col (lane&15).
  #pragma unroll
  for (int mi = 0; mi < 2; ++mi) {
    #pragma unroll
    for (int ni = 0; ni < 4; ++ni) {
      const int gcol = tileN + wn + ni * 16 + mm;
      const float bv = ADD_BIAS ? bias[gcol] : 0.0f;
      #pragma unroll
      for (int r = 0; r < 8; ++r) {
        const int grow = tileM + wm + mi * 16 + r + 8 * hi;
        C[(long long)grow * ldc + gcol] = acc[mi][ni][r] + bv;
      }
    }
  }
}

// In-place softmax over rows of length 2048. One 256-thread block per row,
// 8 elements per thread held in registers across max/sum passes.
__global__ __launch_bounds__(256) void softmax_rows_2048(float* __restrict__ data)
{
  __shared__ float red[256];
  float* p = data + (long long)blockIdx.x * 2048;
  const int tid = threadIdx.x;

  float4 v0 = *(const float4*)(p + tid * 4);
  float4 v1 = *(const float4*)(p + 1024 + tid * 4);

  float m = fmaxf(fmaxf(fmaxf(v0.x, v0.y), fmaxf(v0.z, v0.w)),
                  fmaxf(fmaxf(v1.x, v1.y), fmaxf(v1.z, v1.w)));
  red[tid] = m; __syncthreads();
  for (int s = 128; s > 0; s >>= 1) {
    if (tid < s) red[tid] = fmaxf(red[tid], red[tid + s]);
    __syncthreads();
  }
  m = red[0];
  __syncthreads();

  v0.x = __expf(v0.x - m); v0.y = __expf(v0.y - m);
  v0.z = __expf(v0.z - m); v0.w = __expf(v0.w - m);
  v1.x = __expf(v1.x - m); v1.y = __expf(v1.y - m);
  v1.z = __expf(v1.z - m); v1.w = __expf(v1.w - m);

  float sum = (v0.x + v0.y + v0.z + v0.w) + (v1.x + v1.y + v1.z + v1.w);
  red[tid] = sum; __syncthreads();
  for (int s = 128; s > 0; s >>= 1) {
    if (tid < s) red[tid] += red[tid + s];
    __syncthreads();
  }
  const float inv = 1.0f / red[0];

  v0.x *= inv; v0.y *= inv; v0.z *= inv; v0.w *= inv;
  v1.x *= inv; v1.y *= inv; v1.z *= inv; v1.w *= inv;
  *(float4*)(p + tid * 4) = v0;
  *(float4*)(p + 1024 + tid * 4) = v1;
}

extern "C" void kernel_launch(void* const* d_in, const int* in_sizes, int n_in,
                              void* d_out, int out_size, void* d_ws, size_t ws_size,
                              hipStream_t stream)
{
  (void)in_sizes; (void)n_in; (void)out_size; (void)ws_size;
  const float* dec  = (const float*)d_in[0];   // [16, 2048, 1024]
  const float* enc  = (const float*)d_in[1];   // [16, 2048, 1024]
  const float* W    = (const float*)d_in[2];   // [1024, 1024]
  const float* bias = (const float*)d_in[3];   // [1024]

  constexpr int BZ = 16, TQ = 2048, TK = 2048, DD = 1024;

  float* ctx   = (float*)d_out;                                  // [16, 2048, 1024]
  float* align = (float*)d_out + (long long)BZ * TQ * DD;        // [16, 2048, 2048]
  float* keys  = (float*)d_ws;                                   // [16, 2048, 1024]

  // 1) keys = enc @ W + bias    (per batch: M=TK, N=D, K=D; W shared -> strideB=0)
  wmma_gemm_f32_kernel<false, true>
      <<<dim3(DD / TILE_N, TK / TILE_M, BZ), 256, 0, stream>>>(
      enc, W, bias, keys, DD,
      (long long)TK * DD, 0LL, (long long)TK * DD, DD, DD, DD);

  // 2) score = dec @ keys^T     (M=TQ, N=TK, K=D), written into alignment region
  wmma_gemm_f32_kernel<true, false>
      <<<dim3(TK / TILE_N, TQ / TILE_M, BZ), 256, 0, stream>>>(
      dec, keys, nullptr, align, DD,
      (long long)TQ * DD, (long long)TK * DD, (long long)TQ * TK, DD, DD, TK);

  // 3) alignment = softmax(score) along TK, in place
  softmax_rows_2048<<<dim3(BZ * TQ), 256, 0, stream>>>(align);

  // 4) context = alignment @ enc (M=TQ, N=D, K=TK)
  wmma_gemm_f32_kernel<false, false>
      <<<dim3(DD / TILE_N, TQ / TILE_M, BZ), 256, 0, stream>>>(
      align, enc, nullptr, ctx, TK,
      (long long)TQ * TK, (long long)TK * DD, (long long)TQ * DD, TK, DD, DD);
}